// BahdanauAttention_14663018348638
// MI455X (gfx1250) — compile-verified
//
#include <hip/hip_runtime.h>
#include <math.h>

typedef float v2f __attribute__((ext_vector_type(2)));
typedef float v8f __attribute__((ext_vector_type(8)));

#define BATCH 4
#define TQ    256
#define TK    1024
#define QS    512     // query/key feature dim (also context dim)
#define NATT  128     // attention dim

// gfx1250 hardware tanh (V_TANH_F32) if the toolchain exposes it.
__device__ __forceinline__ float fast_tanh(float x) {
#if __has_builtin(__builtin_amdgcn_tanhf)
  return __builtin_amdgcn_tanhf(x);
#else
  return tanhf(x);
#endif
}

// ---------------------------------------------------------------------------
// Projection GEMM: out(MxN) = A(MxK) @ W(KxN), K=512, N=128, fp32 WMMA 16x16x4.
// One wave per 16x16 output tile, K-loop in steps of 4.
// A layout (16x4 f32): lanes 0-15 hold rows M=0..15 with K={k0,k0+1} in
// vgpr0/1; lanes 16-31 hold K={k0+2,k0+3}.  B (4x16): lane = N, lane-half
// selects K pair (mirrors A).  C/D: vgpr r -> M = r + 8*(lane>=16), N=lane&15.
// TRANS=true stores the result transposed (out[n][m], ld = M) for att_keys.
// ---------------------------------------------------------------------------
template <bool TRANS>
__global__ void proj_kernel(const float* __restrict__ A,
                            const float* __restrict__ W,
                            float* __restrict__ out, int M) {
  const int K = QS, N = NATT;
  int wave = (int)((blockIdx.x * blockDim.x + threadIdx.x) >> 5);
  int lane = (int)(threadIdx.x & 31);
  int ntiles = N / 16;
  int mt = wave / ntiles, nt = wave % ntiles;
  if (mt * 16 >= M) return;

  int row = mt * 16 + (lane & 15);
  int kh  = (lane >> 4) << 1;            // 0 or 2
  int n   = nt * 16 + (lane & 15);

  const float* Arow = A + (size_t)row * K;
  v8f acc = {};
  for (int k0 = 0; k0 < K; k0 += 4) {
    v2f a = *(const v2f*)(Arow + k0 + kh);          // K = k0+kh, k0+kh+1
    v2f b;
    b.x = W[(size_t)(k0 + kh) * N + n];
    b.y = W[(size_t)(k0 + kh + 1) * N + n];
    acc = __builtin_amdgcn_wmma_f32_16x16x4_f32(false, a, false, b,
                                                (short)0, acc, false, false);
  }

  int cn = nt * 16 + (lane & 15);
  int m0 = mt * 16 + ((lane >> 4) << 3);
  if (TRANS) {
#pragma unroll
    for (int r = 0; r < 8; ++r)
      out[(size_t)cn * M + (m0 + r)] = acc[r];
  } else {
#pragma unroll
    for (int r = 0; r < 8; ++r)
      out[(size_t)(m0 + r) * N + cn] = acc[r];
  }
}

// ---------------------------------------------------------------------------
// Fused scores (tanh) + softmax + context.  One block = one (batch, 16-q tile).
// LDS: 16x1024 f32 score tile (64KB) + 16x128 aq tile + v  (~73KB of 320KB).
// ---------------------------------------------------------------------------
__global__ __launch_bounds__(256) void fused_attn(
    const float* __restrict__ aq_all,  // [BATCH*TQ][NATT]
    const float* __restrict__ akT,     // [NATT][BATCH*TK]  (transposed)
    const float* __restrict__ vatt,    // [NATT]
    const float* __restrict__ keys,    // [BATCH][TK][QS]
    float* __restrict__ ctx_out,       // [BATCH][TQ][QS]
    float* __restrict__ w_out)         // [BATCH][TQ][TK]
{
  __shared__ __align__(16) float s_sc[16][TK];
  __shared__ __align__(16) float s_aq[16][NATT];
  __shared__ float s_v[NATT];

  const int tid  = (int)threadIdx.x;
  const int lane = tid & 31;
  const int wav  = tid >> 5;              // 8 waves
  const int g    = (int)blockIdx.x;       // 0..63
  const int b    = g >> 4;                // 16 q-tiles per batch
  const int q0   = (g & 15) * 16;

  // --- stage aq tile and v into LDS (coalesced) ---
  for (int i = tid; i < 16 * NATT; i += 256) {
    int q = i >> 7, nn = i & (NATT - 1);
    s_aq[q][nn] = aq_all[(size_t)(b * TQ + q0 + q) * NATT + nn];
  }
  if (tid < NATT) s_v[tid] = vatt[tid];
  __syncthreads();

  // --- Phase 1: scores[q][k] = sum_n v[n]*tanh(aq[q][n] + ak[k][n]) ---
  const int MK = BATCH * TK;
  for (int c = wav; c < TK / 32; c += 8) {      // 4 chunks of 32 k per wave
    const int kg = b * TK + c * 32 + lane;      // coalesced column of akT
    float s[16];
#pragma unroll
    for (int q = 0; q < 16; ++q) s[q] = 0.f;
    for (int nn = 0; nn < NATT; ++nn) {
      float ak = akT[(size_t)nn * MK + kg];
      float vn = s_v[nn];
#pragma unroll
      for (int q = 0; q < 16; ++q)
        s[q] += vn * fast_tanh(s_aq[q][nn] + ak);
    }
#pragma unroll
    for (int q = 0; q < 16; ++q) s_sc[q][c * 32 + lane] = s[q];
  }
  __syncthreads();

  // --- Phase 2: softmax over k, write weights, keep normalized in LDS ---
  for (int q = wav; q < 16; q += 8) {           // 2 rows per wave
    float m = -1e30f;
    for (int i = lane; i < TK; i += 32) m = fmaxf(m, s_sc[q][i]);
#pragma unroll
    for (int off = 16; off > 0; off >>= 1) m = fmaxf(m, __shfl_xor(m, off, 32));
    float sum = 0.f;
    for (int i = lane; i < TK; i += 32) {
      float e = __expf(s_sc[q][i] - m);
      s_sc[q][i] = e;
      sum += e;
    }
#pragma unroll
    for (int off = 16; off > 0; off >>= 1) sum += __shfl_xor(sum, off, 32);
    float inv = 1.f / sum;
    float* wrow = w_out + (size_t)(b * TQ + q0 + q) * TK;
    for (int i = lane; i < TK; i += 32) {
      float w = s_sc[q][i] * inv;
      s_sc[q][i] = w;
      wrow[i] = w;        // weights output (coalesced)
    }
  }
  __syncthreads();

  // --- Phase 3: context[16][512] = weights[16][1024] @ keys[1024][512] ---
  // fp32 WMMA 16x16x4; A straight out of LDS, B from L2-resident keys.
  const float* kbase = keys + (size_t)b * TK * QS;
  float* crow = ctx_out + (size_t)(b * TQ + q0) * QS;
  const int arow = lane & 15;
  const int khh  = (lane >> 4) << 1;
  for (int nt = wav * 4; nt < wav * 4 + 4; ++nt) {   // 4 d-tiles per wave
    const int dcol = nt * 16 + (lane & 15);
    v8f acc = {};
    for (int k0 = 0; k0 < TK; k0 += 4) {
      v2f a = *(const v2f*)(&s_sc[arow][k0 + khh]);  // 8B-aligned LDS pair
      v2f bb;
      bb.x = kbase[(size_t)(k0 + khh) * QS + dcol];
      bb.y = kbase[(size_t)(k0 + khh + 1) * QS + dcol];
      acc = __builtin_amdgcn_wmma_f32_16x16x4_f32(false, a, false, bb,
                                                  (short)0, acc, false, false);
    }
    const int m0 = (lane >> 4) << 3;
#pragma unroll
    for (int r = 0; r < 8; ++r)
      crow[(size_t)(m0 + r) * QS + dcol] = acc[r];
  }
}

// ---------------------------------------------------------------------------
extern "C" void kernel_launch(void* const* d_in, const int* in_sizes, int n_in,
                              void* d_out, int out_size, void* d_ws,
                              size_t ws_size, hipStream_t stream) {
  const float* query = (const float*)d_in[0];  // [4][256][512]
  const float* keys  = (const float*)d_in[1];  // [4][1024][512]
  const float* Wq    = (const float*)d_in[2];  // [512][128]
  const float* Wk    = (const float*)d_in[3];  // [512][128]
  const float* vatt  = (const float*)d_in[4];  // [128]

  float* ctx = (float*)d_out;                            // [4][256][512]
  float* wts = (float*)d_out + (size_t)BATCH * TQ * QS;  // [4][256][1024]

  float* aq  = (float*)d_ws;                     // [BATCH*TQ][128]
  float* akT = aq + (size_t)BATCH * TQ * NATT;   // [128][BATCH*TK]

  {  // project queries: M = 4*256 = 1024 rows -> 512 waves -> 128 blocks
    int M = BATCH * TQ;
    int waves = (M / 16) * (NATT / 16);
    proj_kernel<false><<<waves / 4, 128, 0, stream>>>(query, Wq, aq, M);
  }
  {  // project keys (transposed output): M = 4*1024 = 4096 rows -> 2048 waves
    int M = BATCH * TK;
    int waves = (M / 16) * (NATT / 16);
    proj_kernel<true><<<waves / 4, 128, 0, stream>>>(keys, Wk, akT, M);
  }
  fused_attn<<<BATCH * TQ / 16, 256, 0, stream>>>(aq, akT, vatt, keys, ctx, wts);
}